// GATLayerOut_single_1932735283952
// MI455X (gfx1250) — compile-verified
//
#include <hip/hip_runtime.h>
#include <cmath>

#define NEG_BIG   (-9000000000000000.0f)
#define LEAKY     0.01f
#define N_NODES   8192
#define F_IN      256
#define F_OUT     128

typedef __attribute__((ext_vector_type(2))) float v2f;
typedef __attribute__((ext_vector_type(8))) float v8f;

// ---------------------------------------------------------------------------
// Kernel 1: h = x @ W     (8192x256 @ 256x128 -> 8192x128), fp32 WMMA 16x16x4
// grid = 512 blocks (one 16-row slab each), block = 256 threads = 8 waves,
// wave w computes column tile [16w, 16w+16).
// ---------------------------------------------------------------------------
__global__ __launch_bounds__(256) void k_h_gemm(const float* __restrict__ x,
                                                const float* __restrict__ W,
                                                float* __restrict__ h) {
  const int wave = threadIdx.x >> 5;
  const int lane = threadIdx.x & 31;
  const int m16  = lane & 15;      // M for A-frag / N for B-frag
  const int kh   = lane >> 4;      // K-half selector
  const int row0 = blockIdx.x * 16;
  const int col0 = wave * 16;

  v8f c = {};
  const float* xrow = x + (size_t)(row0 + m16) * F_IN;

  for (int k = 0; k < F_IN; k += 4) {
    // A: x[row0+m16][k + 2*kh + {0,1}]  (8B aligned)
    v2f a = *(const v2f*)(xrow + k + 2 * kh);
    // B: W[k+2*kh + {0,1}][col0 + m16]
    v2f b;
    b[0] = W[(size_t)(k + 2 * kh) * F_OUT + col0 + m16];
    b[1] = W[(size_t)(k + 2 * kh + 1) * F_OUT + col0 + m16];
    c = __builtin_amdgcn_wmma_f32_16x16x4_f32(false, a, false, b,
                                              (short)0, c, false, false);
  }

  // D layout: VGPR r -> M = r + 8*kh, N = m16
  #pragma unroll
  for (int r = 0; r < 8; ++r)
    h[(size_t)(row0 + r + 8 * kh) * F_OUT + col0 + m16] = c[r];
}

// ---------------------------------------------------------------------------
// Kernel 2: s1[i] = h[i,:]·a[0:128], s2[i] = h[i,:]·a[128:256]
// ---------------------------------------------------------------------------
__global__ __launch_bounds__(256) void k_attn_vec(const float* __restrict__ h,
                                                  const float* __restrict__ a,
                                                  float* __restrict__ s1,
                                                  float* __restrict__ s2) {
  const int i = blockIdx.x * blockDim.x + threadIdx.x;
  if (i >= N_NODES) return;
  const float* hr = h + (size_t)i * F_OUT;
  float acc1 = 0.f, acc2 = 0.f;
  #pragma unroll 8
  for (int f = 0; f < F_OUT; ++f) {
    float hv = hr[f];
    acc1 += hv * a[f];
    acc2 += hv * a[F_OUT + f];
  }
  s1[i] = acc1;
  s2[i] = acc2;
}

// ---------------------------------------------------------------------------
// Kernel 3: per-row online softmax stats (m, l) over
//   score[i,j] = adj[i,j]>0 ? leaky(s1[i]+s2[j]) : NEG_BIG
// one wave per row; exp(NEG_BIG - finite) flushes to 0 (matches f32 reference);
// an all-NEG_BIG row yields l = N (uniform softmax) with no special case.
// ---------------------------------------------------------------------------
__global__ __launch_bounds__(256) void k_row_stats(const int* __restrict__ adj,
                                                   const float* __restrict__ s1,
                                                   const float* __restrict__ s2,
                                                   float* __restrict__ mrow,
                                                   float* __restrict__ lrow) {
  const int wave = threadIdx.x >> 5;
  const int lane = threadIdx.x & 31;
  const int row  = blockIdx.x * 8 + wave;
  const float s1v = s1[row];
  const int* arow = adj + (size_t)row * N_NODES;

  float m = NEG_BIG, l = 0.f;
  for (int j = lane * 2; j < N_NODES; j += 64) {
    int2   ad = *(const int2*)(arow + j);
    float2 sv = *(const float2*)(s2 + j);
    float e0 = s1v + sv.x; e0 = (e0 > 0.f) ? e0 : LEAKY * e0; e0 = (ad.x > 0) ? e0 : NEG_BIG;
    float e1 = s1v + sv.y; e1 = (e1 > 0.f) ? e1 : LEAKY * e1; e1 = (ad.y > 0) ? e1 : NEG_BIG;
    float mp = fmaxf(e0, e1);
    float lp = __expf(e0 - mp) + __expf(e1 - mp);
    float nm = fmaxf(m, mp);
    l = l * __expf(m - nm) + lp * __expf(mp - nm);
    m = nm;
  }
  // merge (m,l) across the 32 lanes
  #pragma unroll
  for (int off = 16; off > 0; off >>= 1) {
    float mo = __shfl_xor(m, off, 32);
    float lo = __shfl_xor(l, off, 32);
    float nm = fmaxf(m, mo);
    l = l * __expf(m - nm) + lo * __expf(mo - nm);
    m = nm;
  }
  if (lane == 0) { mrow[row] = m; lrow[row] = l; }
}

// ---------------------------------------------------------------------------
// Kernel 4: attention write + h1 = elu(attention @ h), fused.
// grid = 512 (one 16-row block each), block = 256 = 8 waves.
// Waves stride the 512 j-blocks; each wave keeps a full 16x128 f32 accumulator
// (8 x v8f) fed by WMMA f32 16x16x4; LDS reduction across waves, then ELU.
// ---------------------------------------------------------------------------
__global__ __launch_bounds__(256) void k_attn_apply(const int* __restrict__ adj,
                                                    const float* __restrict__ h,
                                                    const float* __restrict__ s1,
                                                    const float* __restrict__ s2,
                                                    const float* __restrict__ mrow,
                                                    const float* __restrict__ lrow,
                                                    float* __restrict__ out_h1,
                                                    float* __restrict__ out_att) {
  const int wave = threadIdx.x >> 5;
  const int lane = threadIdx.x & 31;
  const int m16  = lane & 15;
  const int kh   = lane >> 4;
  const int row0 = blockIdx.x * 16;
  const int row  = row0 + m16;

  const float s1v  = s1[row];
  const float mv   = mrow[row];
  const float invl = 1.f / lrow[row];
  const int*   arow = adj     + (size_t)row * N_NODES;
  float*       orow = out_att + (size_t)row * N_NODES;

  v8f acc[8] = {};

  for (int jb = wave; jb < N_NODES / 16; jb += 8) {
    const int j0 = jb * 16;
    __builtin_prefetch(arow + j0 + 128, 0, 1);  // next j-block for this wave

    v2f afrag[4];
    #pragma unroll
    for (int t = 0; t < 4; ++t) {
      const int j = j0 + 4 * t + 2 * kh;
      int2   ad = *(const int2*)(arow + j);
      float2 sv = *(const float2*)(s2 + j);
      float e0 = s1v + sv.x; e0 = (e0 > 0.f) ? e0 : LEAKY * e0; e0 = (ad.x > 0) ? e0 : NEG_BIG;
      float e1 = s1v + sv.y; e1 = (e1 > 0.f) ? e1 : LEAKY * e1; e1 = (ad.y > 0) ? e1 : NEG_BIG;
      float p0 = __expf(e0 - mv) * invl;
      float p1 = __expf(e1 - mv) * invl;
      afrag[t][0] = p0;
      afrag[t][1] = p1;
      *(float2*)(orow + j) = make_float2(p0, p1);
    }

    #pragma unroll
    for (int t = 0; t < 4; ++t) {
      const int jk = j0 + 4 * t;          // B-matrix K rows jk..jk+3
      const float* hb0 = h + (size_t)(jk + 2 * kh) * F_OUT + m16;
      const float* hb1 = hb0 + F_OUT;
      #pragma unroll
      for (int f = 0; f < 8; ++f) {
        v2f b;
        b[0] = hb0[f * 16];
        b[1] = hb1[f * 16];
        acc[f] = __builtin_amdgcn_wmma_f32_16x16x4_f32(false, afrag[t], false, b,
                                                       (short)0, acc[f], false, false);
      }
    }
  }

  // cross-wave reduction of the 16x128 partial tiles in LDS (64 KB)
  __shared__ float red[8][16][F_OUT];
  #pragma unroll
  for (int f = 0; f < 8; ++f)
    #pragma unroll
    for (int r = 0; r < 8; ++r)
      red[wave][r + 8 * kh][f * 16 + m16] = acc[f][r];
  __syncthreads();

  for (int idx = threadIdx.x; idx < 16 * F_OUT; idx += 256) {
    const int rr = idx >> 7;
    const int cc = idx & (F_OUT - 1);
    float v = 0.f;
    #pragma unroll
    for (int w = 0; w < 8; ++w) v += red[w][rr][cc];
    v = (v > 0.f) ? v : expm1f(v);   // ELU, alpha = 1
    out_h1[(size_t)(row0 + rr) * F_OUT + cc] = v;
  }
}

// ---------------------------------------------------------------------------
extern "C" void kernel_launch(void* const* d_in, const int* in_sizes, int n_in,
                              void* d_out, int out_size, void* d_ws, size_t ws_size,
                              hipStream_t stream) {
  const float* x   = (const float*)d_in[0];
  const int*   adj = (const int*)d_in[1];
  const float* W   = (const float*)d_in[2];
  const float* a   = (const float*)d_in[3];

  float* out_h1  = (float*)d_out;                       // 8192*128
  float* out_att = out_h1 + (size_t)N_NODES * F_OUT;    // 8192*8192

  float* h    = (float*)d_ws;                           // 8192*128
  float* s1   = h  + (size_t)N_NODES * F_OUT;           // 8192
  float* s2   = s1 + N_NODES;                           // 8192
  float* mrow = s2 + N_NODES;                           // 8192
  float* lrow = mrow + N_NODES;                         // 8192

  k_h_gemm    <<<N_NODES / 16, 256, 0, stream>>>(x, W, h);
  k_attn_vec  <<<N_NODES / 256, 256, 0, stream>>>(h, a, s1, s2);
  k_row_stats <<<N_NODES / 8, 256, 0, stream>>>(adj, s1, s2, mrow, lrow);
  k_attn_apply<<<N_NODES / 16, 256, 0, stream>>>(adj, h, s1, s2, mrow, lrow,
                                                 out_h1, out_att);
}